// DropNorm_14774687498704
// MI455X (gfx1250) — compile-verified
//
#include <hip/hip_runtime.h>
#include <stdint.h>

#define FEAT   8192
#define BLOCK  256
#define NWAVE  (BLOCK / 32)
#define CHUNKS (FEAT / (BLOCK * 4))   // 8 float4 chunks per thread

typedef float v4f __attribute__((ext_vector_type(4)));

__global__ __launch_bounds__(BLOCK) void dropnorm_kernel(
    const float* __restrict__ x,
    const float* __restrict__ gamma,
    const float* __restrict__ beta,
    const unsigned char* __restrict__ mask,
    float* __restrict__ out)
{
    __shared__ __align__(16) float srow[FEAT];   // one full row staged in LDS (32 KB)
    __shared__ float red[2 * NWAVE + 2];

    const int tid = threadIdx.x;
    const int row = blockIdx.x;
    const float* grow = x + (size_t)row * FEAT;

    // ---- CDNA5 async copy: global row -> LDS (ASYNCcnt path) -------------
    // GVS mode: saddr = row base (SGPR pair), vaddr = per-lane byte offset,
    // vdst = LDS byte address. 256 lanes x 16B x 8 issues = 32 KB, disjoint.
    unsigned lo = (unsigned)(uintptr_t)(&srow[0]) + (unsigned)tid * 16u;
    unsigned go = (unsigned)tid * 16u;
#pragma unroll
    for (int c = 0; c < CHUNKS; ++c) {
        asm volatile("global_load_async_to_lds_b128 %0, %1, %2"
                     :: "v"(lo), "v"(go), "s"(grow)
                     : "memory");
        lo += BLOCK * 16u;
        go += BLOCK * 16u;
    }
#if __has_builtin(__builtin_amdgcn_s_wait_asynccnt)
    __builtin_amdgcn_s_wait_asynccnt(0);
#else
    asm volatile("s_wait_asynccnt 0" ::: "memory");
#endif
    __syncthreads();

    // ---- Pass 1: S1 = sum(m*x), S2 = sum(m*x^2) (single fused pass) ------
    float s1 = 0.f, s2 = 0.f;
    unsigned mw[CHUNKS];                 // keep mask bytes for pass 2
#pragma unroll
    for (int c = 0; c < CHUNKS; ++c) {
        const int f = c * (BLOCK * 4) + tid * 4;
        v4f xv = *(const v4f*)&srow[f];
        unsigned m = *(const unsigned*)(mask + f);   // 4 bool bytes (0/1)
        mw[c] = m;
        const float m0 = (float)(m & 0xFFu);
        const float m1 = (float)((m >> 8)  & 0xFFu);
        const float m2 = (float)((m >> 16) & 0xFFu);
        const float m3 = (float)((m >> 24) & 0xFFu);
        const float t0 = m0 * xv.x, t1 = m1 * xv.y;
        const float t2 = m2 * xv.z, t3 = m3 * xv.w;
        s1 += (t0 + t1) + (t2 + t3);
        s2 = fmaf(t0, xv.x, fmaf(t1, xv.y, fmaf(t2, xv.z, fmaf(t3, xv.w, s2))));
    }

    // ---- wave32 intra-wave reduction, then cross-wave via LDS ------------
#pragma unroll
    for (int off = 16; off > 0; off >>= 1) {
        s1 += __shfl_xor(s1, off, 32);
        s2 += __shfl_xor(s2, off, 32);
    }
    const int wave = tid >> 5;
    if ((tid & 31) == 0) { red[wave] = s1; red[NWAVE + wave] = s2; }
    __syncthreads();
    if (tid == 0) {
        float S1 = 0.f, S2 = 0.f;
#pragma unroll
        for (int w = 0; w < NWAVE; ++w) { S1 += red[w]; S2 += red[NWAVE + w]; }
        const float n      = (float)(FEAT / 2);               // exactly F/2 ones
        const float mu     = S1 * (1.0f / n);
        const float sigma2 = (S2 - n * mu * mu) * (1.0f / (n - 1.0f));
        const float rstd   = rsqrtf(fmaf(sigma2, sigma2, 1e-4f)); // quirk: sigma2^2
        red[2 * NWAVE]     = mu;
        red[2 * NWAVE + 1] = rstd;
    }
    __syncthreads();
    const float mu   = red[2 * NWAVE];
    const float rstd = red[2 * NWAVE + 1];

    // ---- Pass 2: out = gamma * m*(x-mu)*rstd + beta (streamed NT stores) -
    float* orow = out + (size_t)row * FEAT;
#pragma unroll
    for (int c = 0; c < CHUNKS; ++c) {
        const int f = c * (BLOCK * 4) + tid * 4;
        v4f xv = *(const v4f*)&srow[f];
        v4f gv = *(const v4f*)(gamma + f);
        v4f bv = *(const v4f*)(beta + f);
        const unsigned m = mw[c];
        v4f mv;
        mv.x = (float)(m & 0xFFu);
        mv.y = (float)((m >> 8)  & 0xFFu);
        mv.z = (float)((m >> 16) & 0xFFu);
        mv.w = (float)((m >> 24) & 0xFFu);
        v4f xh = (xv - mu) * mv * rstd;
        v4f ov = gv * xh + bv;
        __builtin_nontemporal_store(ov, (v4f*)(orow + f));
    }
}

extern "C" void kernel_launch(void* const* d_in, const int* in_sizes, int n_in,
                              void* d_out, int out_size, void* d_ws, size_t ws_size,
                              hipStream_t stream) {
    const float*         x     = (const float*)d_in[0];
    const float*         gamma = (const float*)d_in[1];
    const float*         beta  = (const float*)d_in[2];
    const unsigned char* mask  = (const unsigned char*)d_in[3];
    float*               out   = (float*)d_out;

    const int B = in_sizes[0] / FEAT;   // 4096 rows
    dropnorm_kernel<<<B, BLOCK, 0, stream>>>(x, gamma, beta, mask, out);
}